// SlotAttention_28527172780279
// MI455X (gfx1250) — compile-verified
//
#include <hip/hip_runtime.h>
#include <math.h>

// ---- problem constants (from reference setup_inputs) ----
constexpr int B_  = 32;
constexpr int N_  = 4096;
constexpr int C_  = 768;
constexpr int D_  = 64;
constexpr int H_  = 128;
constexpr int S_  = 7;
constexpr int MROWS = B_ * N_;   // 131072 rows of x
constexpr int CHUNK = 128;       // n-chunk per attention block
constexpr int ITERS = 3;

typedef float v2f __attribute__((ext_vector_type(2)));
typedef float v8f __attribute__((ext_vector_type(8)));

// =====================================================================
// Kernel 1: xn = LN(x); k = xn@Wk; v = xn@Wv       (fp32 WMMA 16x16x4)
// grid = MROWS/16 blocks, 128 threads (4 waves); each wave: 2 col-tiles
// =====================================================================
__global__ __launch_bounds__(128)
void ln_kv_kernel(const float* __restrict__ x,
                  const float* __restrict__ g_in, const float* __restrict__ b_in,
                  const float* __restrict__ Wk,   const float* __restrict__ Wv,
                  float* __restrict__ kbuf, float* __restrict__ vbuf)
{
    constexpr int PITCH = 772;                 // 768 + 4 -> conflict-free A fetch
    __shared__ float xs[16 * PITCH];           // ~49 KB
    __shared__ float ssum[16], ssumsq[16], smean[16], srstd[16];

    const int tid = threadIdx.x;
    const int rowbase = blockIdx.x * 16;

    // cooperative load: 16 rows x 768 cols, float4
    for (int i = tid; i < 16 * (C_ / 4); i += 128) {
        const int r  = i / (C_ / 4);
        const int c4 = (i % (C_ / 4)) * 4;
        const float4 f = *reinterpret_cast<const float4*>(
            x + (size_t)(rowbase + r) * C_ + c4);
        float* dst = &xs[r * PITCH + c4];
        dst[0] = f.x; dst[1] = f.y; dst[2] = f.z; dst[3] = f.w;
    }
    if (tid < 16) { ssum[tid] = 0.f; ssumsq[tid] = 0.f; }
    __syncthreads();

    // row stats: 8 threads per row, 96 elements each
    {
        const int r = tid >> 3, g = tid & 7;
        float s = 0.f, s2 = 0.f;
        for (int i = 0; i < 96; ++i) {
            const float val = xs[r * PITCH + g * 96 + i];
            s += val; s2 += val * val;
        }
        atomicAdd(&ssum[r], s);
        atomicAdd(&ssumsq[r], s2);
    }
    __syncthreads();
    if (tid < 16) {
        const float mean = ssum[tid] * (1.0f / C_);
        const float var  = ssumsq[tid] * (1.0f / C_) - mean * mean;
        smean[tid] = mean;
        srstd[tid] = rsqrtf(var + 1e-5f);
    }
    __syncthreads();
    {
        const int r = tid >> 3, g = tid & 7;
        const float mean = smean[r], rstd = srstd[r];
        for (int i = 0; i < 96; ++i) {
            const int c = g * 96 + i;
            xs[r * PITCH + c] = (xs[r * PITCH + c] - mean) * rstd * g_in[c] + b_in[c];
        }
    }
    __syncthreads();

    // WMMA GEMM: each wave owns tiles (2w, 2w+1) of 8 (tiles 0-3 -> k, 4-7 -> v)
    const int lane = tid & 31;
    const int wave = tid >> 5;
    const int m    = lane & 15;
    const int koff = (lane >> 4) * 2;          // lanes 16-31 hold K=2,3
    const int t0 = wave * 2, t1 = wave * 2 + 1;
    const float* W0 = (t0 < 4) ? Wk : Wv;
    const float* W1 = (t1 < 4) ? Wk : Wv;
    const int cb0 = (t0 & 3) * 16, cb1 = (t1 & 3) * 16;

    v8f acc0 = {}; v8f acc1 = {};
    for (int kk = 0; kk < C_; kk += 4) {
        v2f a, bv0, bv1;
        a.x = xs[m * PITCH + kk + koff];
        a.y = xs[m * PITCH + kk + koff + 1];
        bv0.x = W0[(size_t)(kk + koff)     * D_ + cb0 + m];
        bv0.y = W0[(size_t)(kk + koff + 1) * D_ + cb0 + m];
        bv1.x = W1[(size_t)(kk + koff)     * D_ + cb1 + m];
        bv1.y = W1[(size_t)(kk + koff + 1) * D_ + cb1 + m];
        acc0 = __builtin_amdgcn_wmma_f32_16x16x4_f32(
            false, a, false, bv0, (short)0, acc0, false, false);
        acc1 = __builtin_amdgcn_wmma_f32_16x16x4_f32(
            false, a, false, bv1, (short)0, acc1, false, false);
    }

    float* out0 = (t0 < 4) ? kbuf : vbuf;
    float* out1 = (t1 < 4) ? kbuf : vbuf;
    const int rplus = (lane >> 4) * 8;         // lanes 16-31 hold rows M+8
#pragma unroll
    for (int j = 0; j < 8; ++j) {
        const int row = rowbase + j + rplus;
        out0[(size_t)row * D_ + cb0 + m] = acc0[j];
        out1[(size_t)row * D_ + cb1 + m] = acc1[j];
    }
}

// =====================================================================
// Kernel 2: q = LN(slots)@Wq; also zero attn_acc / wsum for this iter
// grid = B*S blocks, 64 threads
// =====================================================================
__global__ __launch_bounds__(64)
void q_kernel(const float* __restrict__ slots,
              const float* __restrict__ g_sl, const float* __restrict__ b_sl,
              const float* __restrict__ Wq,
              float* __restrict__ q, float* __restrict__ attn_acc,
              float* __restrict__ wsum)
{
    __shared__ float xn[D_];
    __shared__ float stats[2];
    const int bs = blockIdx.x;
    const int d  = threadIdx.x;

    const float v = slots[(size_t)bs * D_ + d];
    xn[d] = v;
    __syncthreads();
    if (d == 0) {
        float s = 0.f, s2 = 0.f;
        for (int i = 0; i < D_; ++i) { s += xn[i]; s2 += xn[i] * xn[i]; }
        const float mean = s * (1.0f / D_);
        stats[0] = mean;
        stats[1] = rsqrtf(s2 * (1.0f / D_) - mean * mean + 1e-5f);
    }
    __syncthreads();
    xn[d] = (v - stats[0]) * stats[1] * g_sl[d] + b_sl[d];
    __syncthreads();

    float acc = 0.f;
    for (int i = 0; i < D_; ++i) acc += xn[i] * Wq[i * D_ + d];
    q[(size_t)bs * D_ + d]        = acc;
    attn_acc[(size_t)bs * D_ + d] = 0.f;
    if (d == 0) wsum[bs] = 0.f;
}

// =====================================================================
// Kernel 3: logits -> softmax over slots -> partial (w @ v) via WMMA
// grid = (N/CHUNK, B), 128 threads (4 waves); atomics into attn_acc/wsum
// =====================================================================
__global__ __launch_bounds__(128)
void attn_kernel(const float* __restrict__ q, const float* __restrict__ kbuf,
                 const float* __restrict__ vbuf,
                 float* __restrict__ attn_acc, float* __restrict__ wsum,
                 float* __restrict__ weights_out, int store_w)
{
    constexpr int KP = 65;    // k-chunk pitch, conflict-free row reads
    constexpr int WP = 132;   // weight pitch, conflict-free A fetch
    __shared__ float klds[CHUNK * KP];   // ~33 KB
    __shared__ float wlds[16 * WP];      // ~8 KB (rows 7..15 stay zero)
    __shared__ float qs[S_ * D_];

    const int tid   = threadIdx.x;
    const int b     = blockIdx.y;
    const int nbase = blockIdx.x * CHUNK;

    for (int i = tid; i < S_ * D_; i += 128)
        qs[i] = q[(size_t)b * S_ * D_ + i];
    for (int i = tid; i < 16 * WP; i += 128)
        wlds[i] = 0.f;

    const float* ksrc = kbuf + ((size_t)b * N_ + nbase) * D_;
    for (int i = tid; i < CHUNK * (D_ / 4); i += 128) {
        const int r  = i / (D_ / 4);
        const int c4 = (i % (D_ / 4)) * 4;
        const float4 f = *reinterpret_cast<const float4*>(ksrc + (size_t)r * D_ + c4);
        float* dst = &klds[r * KP + c4];
        dst[0] = f.x; dst[1] = f.y; dst[2] = f.z; dst[3] = f.w;
    }
    __syncthreads();

    // logits for this thread's n; softmax over the 7 slots
    float lg[S_];
#pragma unroll
    for (int s = 0; s < S_; ++s) {
        float acc = 0.f;
        for (int i = 0; i < D_; ++i) acc += qs[s * D_ + i] * klds[tid * KP + i];
        lg[s] = acc * 0.125f;                   // scale = D^-0.5
    }
    float mx = lg[0];
#pragma unroll
    for (int s = 1; s < S_; ++s) mx = fmaxf(mx, lg[s]);
    float sum = 0.f;
#pragma unroll
    for (int s = 0; s < S_; ++s) { lg[s] = __expf(lg[s] - mx); sum += lg[s]; }
    const float inv = 1.f / sum;
#pragma unroll
    for (int s = 0; s < S_; ++s) {
        const float w = lg[s] * inv;
        wlds[s * WP + tid] = w;
        if (store_w)
            weights_out[((size_t)b * S_ + s) * N_ + nbase + tid] = w;
    }
    __syncthreads();

    // wsum partials (denominator of weighted mean over N)
    if (tid < S_) {
        float acc = 0.f;
        for (int n = 0; n < CHUNK; ++n) acc += wlds[tid * WP + n];
        atomicAdd(&wsum[b * S_ + tid], acc);
    }

    // partial attn = w(16x128, rows>=7 zero) @ v(128x64) via WMMA f32
    const int lane = tid & 31, wave = tid >> 5;
    const int m    = lane & 15;
    const int koff = (lane >> 4) * 2;
    const int cb   = wave * 16;
    const float* vsrc = vbuf + ((size_t)b * N_ + nbase) * D_;

    v8f acc = {};
    for (int kk = 0; kk < CHUNK; kk += 4) {
        v2f a, bv;
        a.x  = wlds[m * WP + kk + koff];
        a.y  = wlds[m * WP + kk + koff + 1];
        bv.x = vsrc[(size_t)(kk + koff)     * D_ + cb + m];
        bv.y = vsrc[(size_t)(kk + koff + 1) * D_ + cb + m];
        acc = __builtin_amdgcn_wmma_f32_16x16x4_f32(
            false, a, false, bv, (short)0, acc, false, false);
    }
    // lanes 0-15 hold rows M=j (j<8); only slots 0..6 are real
    if (lane < 16) {
#pragma unroll
        for (int j = 0; j < S_; ++j)
            atomicAdd(&attn_acc[((size_t)b * S_ + j) * D_ + cb + m], acc[j]);
    }
}

// =====================================================================
// Kernel 4: attn normalize -> GRUCell -> residual MLP -> new slots
// grid = B*S blocks, 64 threads
// =====================================================================
__global__ __launch_bounds__(64)
void update_kernel(const float* __restrict__ slots_in,
                   const float* __restrict__ attn_acc, const float* __restrict__ wsum,
                   const float* __restrict__ W_ih, const float* __restrict__ W_hh,
                   const float* __restrict__ b_ih, const float* __restrict__ b_hh,
                   const float* __restrict__ g_mlp, const float* __restrict__ b_mlp,
                   const float* __restrict__ w1, const float* __restrict__ bb1,
                   const float* __restrict__ w2, const float* __restrict__ bb2,
                   float* __restrict__ slots_out, float* __restrict__ final_out,
                   int last)
{
    __shared__ float at[D_], sl[D_], hnew[D_], xn[D_], h1[H_];
    __shared__ float stats[2];
    const int bs = blockIdx.x;
    const int d  = threadIdx.x;

    const float denom = wsum[bs] + 1e-8f;
    at[d] = attn_acc[(size_t)bs * D_ + d] / denom;
    const float sv = slots_in[(size_t)bs * D_ + d];
    sl[d] = sv;
    __syncthreads();

    float gi_r = b_ih[d], gi_z = b_ih[D_ + d], gi_n = b_ih[2 * D_ + d];
    float gh_r = b_hh[d], gh_z = b_hh[D_ + d], gh_n = b_hh[2 * D_ + d];
    for (int i = 0; i < D_; ++i) {
        const float ai = at[i], si = sl[i];
        gi_r += ai * W_ih[i * 3 * D_ + d];
        gi_z += ai * W_ih[i * 3 * D_ + D_ + d];
        gi_n += ai * W_ih[i * 3 * D_ + 2 * D_ + d];
        gh_r += si * W_hh[i * 3 * D_ + d];
        gh_z += si * W_hh[i * 3 * D_ + D_ + d];
        gh_n += si * W_hh[i * 3 * D_ + 2 * D_ + d];
    }
    const float r = 1.f / (1.f + __expf(-(gi_r + gh_r)));
    const float z = 1.f / (1.f + __expf(-(gi_z + gh_z)));
    const float n = tanhf(gi_n + r * gh_n);
    const float h = (1.f - z) * n + z * sv;
    hnew[d] = h;
    __syncthreads();

    if (d == 0) {
        float s = 0.f, s2 = 0.f;
        for (int i = 0; i < D_; ++i) { s += hnew[i]; s2 += hnew[i] * hnew[i]; }
        const float mean = s * (1.0f / D_);
        stats[0] = mean;
        stats[1] = rsqrtf(s2 * (1.0f / D_) - mean * mean + 1e-5f);
    }
    __syncthreads();
    xn[d] = (h - stats[0]) * stats[1] * g_mlp[d] + b_mlp[d];
    __syncthreads();

#pragma unroll
    for (int jj = 0; jj < 2; ++jj) {
        const int j = d + jj * D_;
        float acc = bb1[j];
        for (int i = 0; i < D_; ++i) acc += xn[i] * w1[i * H_ + j];
        h1[j] = fmaxf(acc, 0.f);
    }
    __syncthreads();

    float acc = bb2[d];
    for (int jx = 0; jx < H_; ++jx) acc += h1[jx] * w2[jx * D_ + d];
    const float outv = h + acc;
    slots_out[(size_t)bs * D_ + d] = outv;
    if (last) final_out[(size_t)bs * D_ + d] = outv;
}

// =====================================================================
extern "C" void kernel_launch(void* const* d_in, const int* in_sizes, int n_in,
                              void* d_out, int out_size, void* d_ws, size_t ws_size,
                              hipStream_t stream)
{
    (void)in_sizes; (void)n_in; (void)out_size; (void)ws_size;

    const float* x      = (const float*)d_in[0];
    const float* slots0 = (const float*)d_in[1];
    const float* g_in   = (const float*)d_in[2];
    const float* b_in   = (const float*)d_in[3];
    const float* g_sl   = (const float*)d_in[4];
    const float* b_sl   = (const float*)d_in[5];
    const float* g_mlp  = (const float*)d_in[6];
    const float* b_mlp  = (const float*)d_in[7];
    const float* Wq     = (const float*)d_in[8];
    const float* Wk     = (const float*)d_in[9];
    const float* Wv     = (const float*)d_in[10];
    const float* W_ih   = (const float*)d_in[11];
    const float* W_hh   = (const float*)d_in[12];
    const float* b_ih   = (const float*)d_in[13];
    const float* b_hh   = (const float*)d_in[14];
    const float* w1     = (const float*)d_in[15];
    const float* bb1    = (const float*)d_in[16];
    const float* w2     = (const float*)d_in[17];
    const float* bb2    = (const float*)d_in[18];
    // d_in[19] = num_iters (3) — hard-coded; host can't read device mem in capture

    float* out = (float*)d_out;
    float* slots_final = out;                       // B*S*D floats
    float* weights_out = out + B_ * S_ * D_;        // B*S*N floats

    float* ws = (float*)d_ws;
    size_t off = 0;
    float* kbuf     = ws + off; off += (size_t)MROWS * D_;   // 33.5 MB
    float* vbuf     = ws + off; off += (size_t)MROWS * D_;   // 33.5 MB
    float* slots_ws = ws + off; off += (size_t)B_ * S_ * D_;
    float* qbuf     = ws + off; off += (size_t)B_ * S_ * D_;
    float* attn_acc = ws + off; off += (size_t)B_ * S_ * D_;
    float* wsum     = ws + off; off += (size_t)B_ * S_;

    // Phase 1: LN(x) + k/v projections (memory-bound; fp32 WMMA)
    ln_kv_kernel<<<MROWS / 16, 128, 0, stream>>>(x, g_in, b_in, Wk, Wv, kbuf, vbuf);

    // Phase 2: 3 slot-attention iterations
    for (int it = 0; it < ITERS; ++it) {
        const float* sin = (it == 0) ? slots0 : slots_ws;
        const int last = (it == ITERS - 1);
        q_kernel<<<B_ * S_, 64, 0, stream>>>(sin, g_sl, b_sl, Wq,
                                             qbuf, attn_acc, wsum);
        attn_kernel<<<dim3(N_ / CHUNK, B_), 128, 0, stream>>>(
            qbuf, kbuf, vbuf, attn_acc, wsum, weights_out, last);
        update_kernel<<<B_ * S_, 64, 0, stream>>>(
            sin, attn_acc, wsum, W_ih, W_hh, b_ih, b_hh,
            g_mlp, b_mlp, w1, bb1, w2, bb2,
            slots_ws, slots_final, last);
    }
}